// GIN_68616397521286
// MI455X (gfx1250) — compile-verified
//
#include <hip/hip_runtime.h>
#include <stdint.h>

#define N_NODES 100000
#define N_EDGES 1600000

typedef __attribute__((ext_vector_type(2))) float v2f;
typedef __attribute__((ext_vector_type(8))) float v8f;

// ---------------------------------------------------------------------------
// Hardware float atomic add: guaranteed global_atomic_add_f32 either via the
// clang builtin or, if absent, via inline CDNA5 asm (no-return form; tracked
// by STOREcnt, covered by s_endpgm's implicit wait-idle).
// ---------------------------------------------------------------------------
#if defined(__has_builtin)
#if __has_builtin(__builtin_amdgcn_global_atomic_fadd_f32)
#define HAVE_HW_FADD 1
#endif
#endif

__device__ __forceinline__ void atom_add_f32(float* p, float v) {
#ifdef HAVE_HW_FADD
  typedef __attribute__((address_space(1))) float gfloat;
  __builtin_amdgcn_global_atomic_fadd_f32((gfloat*)p, v);
#else
#warning "fadd builtin missing: using inline-asm global_atomic_add_f32"
  asm volatile("global_atomic_add_f32 %0, %1, off"
               :
               : "v"((unsigned long long)(uintptr_t)p), "v"(v)
               : "memory");
#endif
}

// ---------------------------------------------------------------------------
// z <- h  (float4 copy; initializes accumulator with the GIN self-term)
// ---------------------------------------------------------------------------
__global__ void copy_f32x4(const float4* __restrict__ in, float4* __restrict__ out, int n4) {
  int i = blockIdx.x * blockDim.x + threadIdx.x;
  if (i < n4) out[i] = in[i];
}

// ---------------------------------------------------------------------------
// z[dst] += h[src] over all edges. One wave per edge; each lane owns a
// contiguous VEC-float chunk so the gather coalesces into b64/b128 loads.
// Accumulator (51.2 MB) is L2-resident, so atomics run at L2 throughput.
// ---------------------------------------------------------------------------
template <int D>
__global__ void scatter_add_k(const float* __restrict__ H, const int* __restrict__ src,
                              const int* __restrict__ dst, float* __restrict__ Z) {
  constexpr int VEC = D / 32;  // 2 for D=64, 4 for D=128
  int e = blockIdx.x * (blockDim.x >> 5) + (threadIdx.x >> 5);
  if (e >= N_EDGES) return;
  int lane = threadIdx.x & 31;
  int s = src[e];
  int t = dst[e];
  const float* hs = H + (size_t)s * D + lane * VEC;
  float* zt = Z + (size_t)t * D + lane * VEC;
  float r[VEC];
#pragma unroll
  for (int i = 0; i < VEC; ++i) r[i] = hs[i];  // merges into vector load
#pragma unroll
  for (int i = 0; i < VEC; ++i) atom_add_f32(&zt[i], r[i]);
}

// ---------------------------------------------------------------------------
// OUT[16 x NOUT tile] = act(Z[16 x K] @ W[K x NOUT] + bias) via fp32 WMMA.
// Optionally dual-writes the result to ZNEXT (next layer's accumulator),
// fusing the "z <- h" init pass into the GEMM store. ZNEXT may equal Z:
// each block stages its 16 rows into LDS before any global store, and only
// ever writes the rows it read, so self-overwrite is race-free.
//
// V_WMMA_F32_16X16X4_F32 operand layouts (ISA 7.12.2):
//   A 16x4 : lane L holds row M=L%16; vgpr v holds K = v + 2*(L/16)
//   B 4x16 : lane L holds col N=L%16; vgpr v holds K = v + 2*(L/16)
//   C 16x16: vgpr r holds M = r + 8*(L/16), N = L%16
// ---------------------------------------------------------------------------
template <int K, int NOUT, bool RELU>
__global__ __launch_bounds__(256) void gemm_wmma_f32(const float* Z,
                                                     const float* __restrict__ W,
                                                     const float* __restrict__ bias,
                                                     float* OUT, float* ZNEXT) {
  __shared__ float As[16 * (K + 1)];
  const int node0 = blockIdx.x * 16;

  // Cooperative, coalesced load of the 16 x K A-tile into LDS.
  for (int i = threadIdx.x; i < 16 * K; i += 256) {
    int r = i / K;
    int c = i - r * K;
    As[r * (K + 1) + c] = Z[(size_t)(node0 + r) * K + c];
  }
  __syncthreads();

  const int wv = threadIdx.x >> 5;      // wave id within block
  constexpr int NT = (NOUT + 15) / 16;  // number of 16-wide column tiles
  if (wv < NT) {                        // wave-uniform: EXEC all-ones inside
    const int lane = threadIdx.x & 31;
    const int hh = lane >> 4;           // lane-half select
    const int lr = lane & 15;
    const int n = wv * 16 + lr;         // output column
    const bool nok = (n < NOUT);

    v8f c = {0.f, 0.f, 0.f, 0.f, 0.f, 0.f, 0.f, 0.f};
#pragma unroll
    for (int k0 = 0; k0 < K; k0 += 4) {
      const int ka = k0 + 2 * hh;
      v2f a, b;
      a.x = As[lr * (K + 1) + ka];
      a.y = As[lr * (K + 1) + ka + 1];
      b.x = nok ? W[(size_t)ka * NOUT + n] : 0.f;
      b.y = nok ? W[(size_t)(ka + 1) * NOUT + n] : 0.f;
      c = __builtin_amdgcn_wmma_f32_16x16x4_f32(false, a, false, b, (short)0, c, false, false);
    }

    if (nok) {
      const float bv = bias[n];
#pragma unroll
      for (int r = 0; r < 8; ++r) {
        const int m = r + 8 * hh;
        float v = c[r] + bv;
        if (RELU) v = (v > 0.f) ? v : 0.f;
        const size_t off = (size_t)(node0 + m) * NOUT + n;
        OUT[off] = v;
        if (ZNEXT) ZNEXT[off] = v;
      }
    }
  }
}

// ---------------------------------------------------------------------------
// Launcher: 3 GIN layers. Picks fused-init plan (128 MB ws) when workspace
// allows, else the copy-based plan (102.4 MB ws). Not bitwise deterministic (fp32 atomics).
// ---------------------------------------------------------------------------
extern "C" void kernel_launch(void* const* d_in, const int* in_sizes, int n_in,
                              void* d_out, int out_size, void* d_ws, size_t ws_size,
                              hipStream_t stream) {
  (void)in_sizes; (void)n_in; (void)out_size;
  const float* x  = (const float*)d_in[0];
  const int*   ei = (const int*)d_in[1];
  const float* W1 = (const float*)d_in[2];
  const float* b1 = (const float*)d_in[3];
  const float* W2 = (const float*)d_in[4];
  const float* b2 = (const float*)d_in[5];
  const float* W3 = (const float*)d_in[6];
  const float* b3 = (const float*)d_in[7];
  const int* src = ei;
  const int* dst = ei + N_EDGES;
  float* out = (float*)d_out;

  const dim3 blk(256);
  const int scat_blocks = N_EDGES / 8;   // 8 waves (edges) per 256-thread block
  const int gemm_blocks = N_NODES / 16;  // 6250 exactly

  const size_t SZ64  = (size_t)N_NODES * 64;   // floats
  const size_t SZ128 = (size_t)N_NODES * 128;  // floats

  if (ws_size >= (SZ64 + 2 * SZ128) * sizeof(float)) {
    // ---- Fused plan: z1 | z2 | h regions (128 MB) ----
    float* z1 = (float*)d_ws;       // 100000 x 64
    float* z2 = z1 + SZ64;          // 100000 x 128 (doubles as z3)
    float* h  = z2 + SZ128;         // 100000 x 128

    // Layer 1: 64 -> 128, relu. h1 -> h and z2 (accumulator init).
    copy_f32x4<<<(int)(SZ64 / 4 + 255) / 256, blk, 0, stream>>>((const float4*)x, (float4*)z1,
                                                                (int)(SZ64 / 4));
    scatter_add_k<64><<<scat_blocks, blk, 0, stream>>>(x, src, dst, z1);
    gemm_wmma_f32<64, 128, true><<<gemm_blocks, blk, 0, stream>>>(z1, W1, b1, h, z2);

    // Layer 2: 128 -> 128, relu. h2 -> h (over h1), z3 init -> z2 itself.
    scatter_add_k<128><<<scat_blocks, blk, 0, stream>>>(h, src, dst, z2);
    gemm_wmma_f32<128, 128, true><<<gemm_blocks, blk, 0, stream>>>(z2, W2, b2, h, z2);

    // Layer 3: 128 -> 2, no relu.
    scatter_add_k<128><<<scat_blocks, blk, 0, stream>>>(h, src, dst, z2);
    gemm_wmma_f32<128, 2, false><<<gemm_blocks, blk, 0, stream>>>(z2, W3, b3, out, nullptr);
  } else {
    // ---- Copy-based plan: zA | hB regions (102.4 MB) ----
    float* zA = (float*)d_ws;
    float* hB = zA + SZ128;

    copy_f32x4<<<(int)(SZ64 / 4 + 255) / 256, blk, 0, stream>>>((const float4*)x, (float4*)zA,
                                                                (int)(SZ64 / 4));
    scatter_add_k<64><<<scat_blocks, blk, 0, stream>>>(x, src, dst, zA);
    gemm_wmma_f32<64, 128, true><<<gemm_blocks, blk, 0, stream>>>(zA, W1, b1, hB, nullptr);

    copy_f32x4<<<(int)(SZ128 / 4 + 255) / 256, blk, 0, stream>>>((const float4*)hB, (float4*)zA,
                                                                 (int)(SZ128 / 4));
    scatter_add_k<128><<<scat_blocks, blk, 0, stream>>>(hB, src, dst, zA);
    gemm_wmma_f32<128, 128, true><<<gemm_blocks, blk, 0, stream>>>(zA, W2, b2, hB, nullptr);

    copy_f32x4<<<(int)(SZ128 / 4 + 255) / 256, blk, 0, stream>>>((const float4*)hB, (float4*)zA,
                                                                 (int)(SZ128 / 4));
    scatter_add_k<128><<<scat_blocks, blk, 0, stream>>>(hB, src, dst, zA);
    gemm_wmma_f32<128, 2, false><<<gemm_blocks, blk, 0, stream>>>(zA, W3, b3, out, nullptr);
  }
}